// MRF_29463475650829
// MI455X (gfx1250) — compile-verified
//
#include <hip/hip_runtime.h>
#include <hip/hip_bf16.h>
#include <math.h>

typedef __attribute__((ext_vector_type(2))) float v2f;
typedef __attribute__((ext_vector_type(8))) float v8f;

#define MRF_B 32
#define MRF_L 256
#define NAA   20
#define LAM_S 1.0f
#define LAM_P 51.0f               // 0.2 * (L-1)
#define BLKPAD 648                // 32 rows x 20 cols padded to 648 (16B-aligned stride)

// One workgroup per residue position i. Streams the 410 KB row
// wp[(i*L+p)*400 ...] exactly once through double-buffered LDS and fuses:
//   - pair regularizer partial   (load threads)
//   - energy[b,i] gather-sum     (wave 0, lane = b)
//   - batch-independent logits   (wave 1, p <= i terms)
//   - batch-dependent logits via one-hot f32 WMMA (waves 4..7)
__global__ __launch_bounds__(256) void mrf_row_kernel(
    const int*   __restrict__ x,
    const float* __restrict__ ws,
    const float* __restrict__ wp,
    float*       __restrict__ partials)
{
    __shared__ int   xT[MRF_L * MRF_B];      // transposed: xT[p*32 + b] (conflict-free)
    __shared__ float buf[2][BLKPAD];         // 400 live floats + zero pad rows 20..31
    __shared__ float logits[MRF_B][24];      // WMMA output staging
    __shared__ float lc[NAA];                // batch-independent logit terms
    __shared__ float red[256];

    const int i    = blockIdx.x;
    const int tid  = threadIdx.x;
    const int lane = tid & 31;
    const int wave = tid >> 5;

    // Preload all of x (32 KB) transposed into LDS.
    for (int t = tid; t < MRF_B * MRF_L; t += 256) {
        const int b = t >> 8;                // t / L
        const int p = t & (MRF_L - 1);
        xT[p * MRF_B + b] = x[t];
    }
    for (int t = tid; t < MRF_B * 24; t += 256) ((float*)logits)[t] = 0.0f;
    // Zero the pad region once: rows j = 20..31 stay zero forever, so the
    // WMMA A-operand loads need no bounds predication (no EXEC save/restore).
    if (tid < BLKPAD - 400) {
        buf[0][400 + tid] = 0.0f;
        buf[1][400 + tid] = 0.0f;
    }
    __syncthreads();

    const float* __restrict__ wprow = wp + (size_t)i * MRF_L * (NAA * NAA);

    float reg_acc = 0.0f;                    // load threads: pair regularizer
    float e_acc   = 0.0f;                    // wave 0: energy, lane = batch
    float lc_reg  = 0.0f;                    // wave 1: lane = j

    // WMMA tiling for waves 4..7: D[j, b] tiles of 16x16, K = one-hot over aa.
    const int wid  = wave - 4;
    const int jt   = (wid >> 1) & 1;         // j-tile (j = 0..15 / 16..31, rows >=20 are pad)
    const int bt   = wid & 1;                // batch-tile
    const int jrow = jt * 16 + (lane & 15);  // A: M = j
    const int klow = (lane < 16) ? 0 : 2;    // f32 A/B K layout: v0={K0,K2}, v1={K1,K3}
    const int bcol = bt * 16 + (lane & 15);  // B/D: N = batch
    v8f acc = {0.f, 0.f, 0.f, 0.f, 0.f, 0.f, 0.f, 0.f};

    const int xi = xT[i * MRF_B + lane];     // x[lane, i]

    // Prologue: load block p = 0 as 100 x float4 (global_load_b128).
    if (tid < 100) {
        const float4 v = ((const float4*)wprow)[tid];
        ((float4*)&buf[0][0])[tid] = v;
        reg_acc += v.x * v.x + fabsf(v.x) + v.y * v.y + fabsf(v.y)
                 + v.z * v.z + fabsf(v.z) + v.w * v.w + fabsf(v.w);
    }
    __syncthreads();

    for (int p = 0; p < MRF_L; ++p) {
        const int cur = p & 1;
        if (p + 1 < MRF_L && tid < 100) {    // fill other buffer (b128 path)
            const float4 v = ((const float4*)(wprow + (p + 1) * 400))[tid];
            ((float4*)&buf[cur ^ 1][0])[tid] = v;
            reg_acc += v.x * v.x + fabsf(v.x) + v.y * v.y + fabsf(v.y)
                     + v.z * v.z + fabsf(v.z) + v.w * v.w + fabsf(v.w);
        }
        if (p + 2 < MRF_L && tid >= 128 && tid < 141)  // 1600 B = 13 lines, 2 blocks ahead
            __builtin_prefetch(wprow + (p + 2) * 400 + (tid - 128) * 32, 0, 0);

        const float* blk = buf[cur];

        if (wave == 0) {
            // energy[b,i] += wp[..., x[b,i], x[b,p]]
            e_acc += blk[xi * NAA + xT[p * MRF_B + lane]];
        } else if (wave == 1) {
            if (lane < NAA) {
                if (p < i)       lc_reg += blk[lane * NAA + (NAA - 1)]; // eff = 19
                else if (p == i) lc_reg += blk[lane * NAA + lane];       // eff = j
            }
        } else if (wave >= 4 && p > i) {
            // D[j,b] += sum_a blk[j*20+a] * onehot(x[b,p] == a), exact in f32.
            const int xbp = xT[p * MRF_B + bcol];
            v2f Am[5];
            #pragma unroll
            for (int c = 0; c < 5; ++c)      // hoisted ds_load_b64s (index is even -> 8B aligned)
                Am[c] = *(const v2f*)(blk + jrow * NAA + c * 4 + klow);
            #pragma unroll
            for (int c = 0; c < 5; ++c) {    // 5 back-to-back WMMAs, one dscnt wait
                v2f Bm;
                Bm.x = (xbp == c * 4 + klow)     ? 1.0f : 0.0f;
                Bm.y = (xbp == c * 4 + klow + 1) ? 1.0f : 0.0f;
                acc = __builtin_amdgcn_wmma_f32_16x16x4_f32(
                    false, Am[c], false, Bm, (short)0, acc, false, false);
            }
        }
        __syncthreads();
    }

    // Spill WMMA accumulators: D layout VGPR v -> M = v (lanes 0-15) / v+8 (16-31).
    if (wave >= 4) {
        #pragma unroll
        for (int v = 0; v < 8; ++v) {
            const int j = jt * 16 + v + ((lane < 16) ? 0 : 8);
            if (j < NAA) logits[bcol][j] = acc[v];
        }
    }
    if (wave == 1 && lane < NAA) lc[lane] = lc_reg;

    // Fixed-order block reduction of the pair-regularizer partial.
    red[tid] = reg_acc;
    __syncthreads();
    #pragma unroll
    for (int s = 128; s > 0; s >>= 1) {
        if (tid < s) red[tid] += red[tid + s];
        __syncthreads();
    }
    if (tid == 0) partials[MRF_L + i] = red[0] * LAM_P;

    // Wave 0: logsumexp over j, subtract energy, reduce over batches.
    if (wave == 0) {
        const int b = lane;
        float lv[NAA];
        float mx = -3.4e38f;
        #pragma unroll
        for (int j = 0; j < NAA; ++j) {
            lv[j] = ws[i * NAA + j] + lc[j] + logits[b][j];
            mx = fmaxf(mx, lv[j]);
        }
        float sum = 0.0f;
        #pragma unroll
        for (int j = 0; j < NAA; ++j) sum += expf(lv[j] - mx);
        const float lse = mx + logf(sum);
        float val = lse - (ws[i * NAA + xi] + e_acc);
        #pragma unroll
        for (int off = 16; off > 0; off >>= 1)
            val += __shfl_down(val, off, 32);
        if (lane == 0) partials[i] = val;
    }
}

// Single block: w_single regularizer + deterministic fixed-order reduction
// of the 512 per-row partials (256 energy sums, 256 pair-reg sums).
__global__ __launch_bounds__(256) void mrf_final_kernel(
    const float* __restrict__ w_single,
    const float* __restrict__ partials,
    float*       __restrict__ out)
{
    __shared__ float red[256];
    const int tid = threadIdx.x;
    float acc = 0.0f;
    for (int t = tid; t < MRF_L * NAA; t += 256) {
        const float v = w_single[t];
        acc += LAM_S * (v * v + fabsf(v));
    }
    acc += partials[tid] * (1.0f / (float)MRF_B);  // mean over batch
    acc += partials[MRF_L + tid];                  // pair regularizer
    red[tid] = acc;
    __syncthreads();
    #pragma unroll
    for (int s = 128; s > 0; s >>= 1) {
        if (tid < s) red[tid] += red[tid + s];
        __syncthreads();
    }
    if (tid == 0) out[0] = red[0];
}

extern "C" void kernel_launch(void* const* d_in, const int* in_sizes, int n_in,
                              void* d_out, int out_size, void* d_ws, size_t ws_size,
                              hipStream_t stream) {
    const int*   x  = (const int*)d_in[0];     // (32, 256) int32
    const float* ws = (const float*)d_in[1];   // (5120, 1) f32
    const float* wp = (const float*)d_in[2];   // (26214400, 1) f32
    float* out      = (float*)d_out;
    float* partials = (float*)d_ws;            // 512 floats used

    mrf_row_kernel<<<MRF_L, 256, 0, stream>>>(x, ws, wp, partials);
    mrf_final_kernel<<<1, 256, 0, stream>>>(ws, partials, out);
}